// MoEFeedForward_7722351198651
// MI455X (gfx1250) — compile-verified
//
#include <hip/hip_runtime.h>
#include <hip/hip_bf16.h>
#include <math.h>

#define TOKENS 64
#define DIM    512
#define HIDDEN 1024
#define TWO_H  2048
#define NEXP   32
#define TOPK   4
#define LIMIT  7.0f
#define ALPHA  1.702f

typedef float v2f __attribute__((ext_vector_type(2)));
typedef float v8f __attribute__((ext_vector_type(8)));

// ---------------- Kernel 1: router (logits -> top4 -> softmax) ---------------
__global__ __launch_bounds__(64) void router_kernel(
    const float* __restrict__ x, const float* __restrict__ Wg,
    const float* __restrict__ bg, int* __restrict__ topk_idx,
    float* __restrict__ topk_w) {
  int t = threadIdx.x;
  if (t >= TOKENS) return;
  const float* xr = x + t * DIM;
  float logits[NEXP];
  for (int e = 0; e < NEXP; ++e) {
    const float* wr = Wg + e * DIM;
    float s = bg[e];
#pragma unroll 8
    for (int k = 0; k < DIM; ++k) s += xr[k] * wr[k];
    logits[e] = s;
  }
  int idx4[TOPK]; float v4[TOPK];
  unsigned used = 0u;
  for (int s = 0; s < TOPK; ++s) {
    float best = -1e30f; int bi = 0;
    for (int e = 0; e < NEXP; ++e)
      if (!((used >> e) & 1u) && logits[e] > best) { best = logits[e]; bi = e; }
    used |= 1u << bi; idx4[s] = bi; v4[s] = best;
  }
  float m = v4[0], den = 0.f, ex[TOPK];
  for (int s = 0; s < TOPK; ++s) { ex[s] = __expf(v4[s] - m); den += ex[s]; }
  for (int s = 0; s < TOPK; ++s) {
    topk_idx[t * TOPK + s] = idx4[s];
    topk_w[t * TOPK + s]   = ex[s] / den;
  }
}

// ------------- Kernel 2: deterministic per-expert token grouping -------------
// lists[e][j] = pid (= token*4 + slot) in ascending pid order; pad with 0.
__global__ __launch_bounds__(32) void group_kernel(
    const int* __restrict__ topk_idx, int* __restrict__ cnt,
    int* __restrict__ lists) {
  int e = threadIdx.x;
  if (e >= NEXP) return;
  int c = 0;
  for (int p = 0; p < TOKENS * TOPK; ++p)
    if (topk_idx[p] == e) lists[e * TOKENS + c++] = p;
  cnt[e] = c;
  for (int j = c; j < TOKENS; ++j) lists[e * TOKENS + j] = 0;
}

// ------- Kernel 3: GEMM1 (x @ w1[e]^T + b1) fused with interleaved SwiGLU ----
// grid = (NEXP, TWO_H/64); block = 128 (4 waves); wave owns 16 output channels
// and always accumulates 4 M-tiles (64 padded token slots) so the k-loop is
// branch-free straight-line code: 1 B-load + 4 A-loads + 4 WMMAs per k-step.
__global__ __launch_bounds__(128) void gemm1_swiglu_kernel(
    const float* __restrict__ x, const float* __restrict__ w1,
    const float* __restrict__ b1, const int* __restrict__ cnt,
    const int* __restrict__ lists, float* __restrict__ h_ws) {
  const int e = blockIdx.x;
  const int c = cnt[e];
  if (c == 0) return;                         // block-uniform: EXEC stays full
  const int lane = threadIdx.x & 31;
  const int wave = threadIdx.x >> 5;
  const int npos = lane & 15;
  const int half = lane >> 4;
  const int nbase = blockIdx.y * 64 + wave * 16;
  const int col   = nbase + npos;             // channel in [0, 2H)
  const int koff  = half * 2;                 // K offset per A/B layout

  const int* lst = lists + e * TOKENS;
  const float* a0 = x + (size_t)(lst[ 0 + npos] >> 2) * DIM;
  const float* a1 = x + (size_t)(lst[16 + npos] >> 2) * DIM;
  const float* a2 = x + (size_t)(lst[32 + npos] >> 2) * DIM;
  const float* a3 = x + (size_t)(lst[48 + npos] >> 2) * DIM;
  const float* brow = w1 + (size_t)e * TWO_H * DIM + (size_t)col * DIM;

  v8f acc[4] = {};
#pragma unroll 2
  for (int k0 = 0; k0 < DIM; k0 += 4) {
    __builtin_prefetch(brow + k0 + 128, 0, 0);   // global_prefetch_b8 ahead
    v2f b  = *(const v2f*)(brow + k0 + koff);
    v2f f0 = *(const v2f*)(a0 + k0 + koff);
    v2f f1 = *(const v2f*)(a1 + k0 + koff);
    v2f f2 = *(const v2f*)(a2 + k0 + koff);
    v2f f3 = *(const v2f*)(a3 + k0 + koff);
    acc[0] = __builtin_amdgcn_wmma_f32_16x16x4_f32(false, f0, false, b,
                                                   (short)0, acc[0], false, false);
    acc[1] = __builtin_amdgcn_wmma_f32_16x16x4_f32(false, f1, false, b,
                                                   (short)0, acc[1], false, false);
    acc[2] = __builtin_amdgcn_wmma_f32_16x16x4_f32(false, f2, false, b,
                                                   (short)0, acc[2], false, false);
    acc[3] = __builtin_amdgcn_wmma_f32_16x16x4_f32(false, f3, false, b,
                                                   (short)0, acc[3], false, false);
  }

  // D layout: lane L, vgpr v -> row M = v + 8*(L/16), col N = L%16.
  // SwiGLU pairs (2i, 2i+1) sit on adjacent lanes of the same VGPR.
  const float bb = b1[e * TWO_H + col];
#pragma unroll
  for (int mt = 0; mt < 4; ++mt) {
#pragma unroll
    for (int v = 0; v < 8; ++v) {
      int m = mt * 16 + half * 8 + v;
      float val   = acc[mt][v] + bb;
      float other = __shfl_xor(val, 1);
      if (((lane & 1) == 0) && m < c) {
        int pid = lst[m];
        float glu = fminf(val, LIMIT);
        float lin = fminf(fmaxf(other, -LIMIT), LIMIT);
        float sig = 1.0f / (1.0f + __expf(-ALPHA * glu));
        h_ws[(size_t)pid * HIDDEN + (col >> 1)] = glu * sig * (lin + 1.0f);
      }
    }
  }
}

// -------- Kernel 4: GEMM2 (h @ w2[e]^T + b2) -> per-(token,slot) rows --------
// grid = (NEXP, DIM/64); block = 128 (4 waves). Same branch-free structure.
__global__ __launch_bounds__(128) void gemm2_kernel(
    const float* __restrict__ h_ws, const float* __restrict__ w2,
    const float* __restrict__ b2, const int* __restrict__ cnt,
    const int* __restrict__ lists, float* __restrict__ y_ws) {
  const int e = blockIdx.x;
  const int c = cnt[e];
  if (c == 0) return;
  const int lane = threadIdx.x & 31;
  const int wave = threadIdx.x >> 5;
  const int npos = lane & 15;
  const int half = lane >> 4;
  const int nbase = blockIdx.y * 64 + wave * 16;
  const int col   = nbase + npos;             // output dim in [0, 512)
  const int koff  = half * 2;

  const int* lst = lists + e * TOKENS;
  const float* a0 = h_ws + (size_t)lst[ 0 + npos] * HIDDEN;
  const float* a1 = h_ws + (size_t)lst[16 + npos] * HIDDEN;
  const float* a2 = h_ws + (size_t)lst[32 + npos] * HIDDEN;
  const float* a3 = h_ws + (size_t)lst[48 + npos] * HIDDEN;
  const float* brow = w2 + (size_t)e * DIM * HIDDEN + (size_t)col * HIDDEN;

  v8f acc[4] = {};
#pragma unroll 2
  for (int k0 = 0; k0 < HIDDEN; k0 += 4) {
    __builtin_prefetch(brow + k0 + 128, 0, 0);
    v2f b  = *(const v2f*)(brow + k0 + koff);
    v2f f0 = *(const v2f*)(a0 + k0 + koff);
    v2f f1 = *(const v2f*)(a1 + k0 + koff);
    v2f f2 = *(const v2f*)(a2 + k0 + koff);
    v2f f3 = *(const v2f*)(a3 + k0 + koff);
    acc[0] = __builtin_amdgcn_wmma_f32_16x16x4_f32(false, f0, false, b,
                                                   (short)0, acc[0], false, false);
    acc[1] = __builtin_amdgcn_wmma_f32_16x16x4_f32(false, f1, false, b,
                                                   (short)0, acc[1], false, false);
    acc[2] = __builtin_amdgcn_wmma_f32_16x16x4_f32(false, f2, false, b,
                                                   (short)0, acc[2], false, false);
    acc[3] = __builtin_amdgcn_wmma_f32_16x16x4_f32(false, f3, false, b,
                                                   (short)0, acc[3], false, false);
  }

  const float bb = b2[e * DIM + col];
#pragma unroll
  for (int mt = 0; mt < 4; ++mt) {
#pragma unroll
    for (int v = 0; v < 8; ++v) {
      int m = mt * 16 + half * 8 + v;
      if (m < c) {
        int pid = lst[m];
        y_ws[(size_t)pid * DIM + col] = acc[mt][v] + bb;
      }
    }
  }
}

// -------- Kernel 5: combine: out[t] = sum_s ew[t][s] * y[t][s] ---------------
__global__ __launch_bounds__(256) void combine_kernel(
    const float* __restrict__ y_ws, const float* __restrict__ topk_w,
    float* __restrict__ out) {
  int i = blockIdx.x * blockDim.x + threadIdx.x;
  if (i >= TOKENS * DIM) return;
  int t = i >> 9;             // / DIM
  int d = i & (DIM - 1);
  float s = 0.f;
#pragma unroll
  for (int k = 0; k < TOPK; ++k)
    s += topk_w[t * TOPK + k] * y_ws[(size_t)(t * TOPK + k) * DIM + d];
  out[i] = s;
}

extern "C" void kernel_launch(void* const* d_in, const int* in_sizes, int n_in,
                              void* d_out, int out_size, void* d_ws, size_t ws_size,
                              hipStream_t stream) {
  const float* x  = (const float*)d_in[0];
  const float* Wg = (const float*)d_in[1];
  const float* bg = (const float*)d_in[2];
  const float* w1 = (const float*)d_in[3];
  const float* b1 = (const float*)d_in[4];
  const float* w2 = (const float*)d_in[5];
  const float* b2 = (const float*)d_in[6];
  float* out = (float*)d_out;

  // Workspace layout (~1.6 MB total)
  char*  ws       = (char*)d_ws;
  int*   topk_idx = (int*)(ws);                             // 256 ints
  float* topk_w   = (float*)(ws + 1024);                    // 256 floats
  int*   cnt      = (int*)(ws + 2048);                      // 32 ints
  int*   lists    = (int*)(ws + 4096);                      // 32*64 ints
  float* h_ws     = (float*)(ws + 16384);                   // 256*1024 floats
  float* y_ws     = (float*)(ws + 16384 + 256 * 1024 * 4);  // 256*512 floats

  router_kernel<<<1, 64, 0, stream>>>(x, Wg, bg, topk_idx, topk_w);
  group_kernel<<<1, 32, 0, stream>>>(topk_idx, cnt, lists);
  gemm1_swiglu_kernel<<<dim3(NEXP, TWO_H / 64), 128, 0, stream>>>(
      x, w1, b1, cnt, lists, h_ws);
  gemm2_kernel<<<dim3(NEXP, DIM / 64), 128, 0, stream>>>(
      h_ws, w2, b2, cnt, lists, y_ws);
  combine_kernel<<<(TOKENS * DIM + 255) / 256, 256, 0, stream>>>(
      y_ws, topk_w, out);
}